// Black_oil_59682865545830
// MI455X (gfx1250) — compile-verified
//
#include <hip/hip_runtime.h>
#include <stdint.h>

// Problem constants (from reference)
#define NXc   128
#define NYc   128
#define NZc   4
#define Bc    8
#define Tc    10
#define PLANE (NXc * NYc)          // 16384 elements per (b,t,z) plane
#define TILEX 32                   // rows per block
#define HROWS (TILEX + 2)          // +1 halo row each side
#define NTILE (NXc / TILEX)        // 4 tiles per plane

typedef __attribute__((ext_vector_type(4))) float v4f;   // native vector: OK for NT builtins

__global__ __launch_bounds__(256)
void blackoil_pino_kernel(const float* __restrict__ pressure,
                          const float* __restrict__ perm,
                          const float* __restrict__ Q,
                          const float* __restrict__ Qw,
                          const float* __restrict__ Timef,
                          const float* __restrict__ Phi,
                          const float* __restrict__ Swini,
                          const float* __restrict__ sat,
                          float* __restrict__ p_loss,
                          float* __restrict__ s_loss)
{
    // LDS staging: pressure tile (with x-halo) + t=0 perm tile (with x-halo)
    __shared__ __attribute__((aligned(16))) float sP [HROWS * NYc];  // 17408 B
    __shared__ __attribute__((aligned(16))) float sK0[HROWS * NYc];  // 17408 B

    const int tile = blockIdx.x & (NTILE - 1);
    const int slab = blockIdx.x >> 2;             // ((b*T)+t)*NZ + z , 0..319
    const int z  = slab % NZc;
    const int bt = slab / NZc;
    const int t  = bt % Tc;
    const int b  = bt / Tc;
    const int x0 = tile * TILEX;

    const int base  = slab * PLANE;                       // this (b,t,z) plane
    const int base0 = (b * Tc * NZc + z) * PLANE;         // (b, t=0, z) plane

    // ---- Stage 1: async copy the two stencil planes into LDS -------------
    // One wave row-job = 128 floats = 512 B = one b128 async instruction.
    const int lane = threadIdx.x & 31;
    const int wave = threadIdx.x >> 5;
    for (int j = wave; j < 2 * HROWS; j += 8) {
        const int  sel = (j >= HROWS);                    // 0 = pressure, 1 = perm(t=0)
        const int  r   = sel ? (j - HROWS) : j;           // LDS row 0..33
        int grow = x0 - 1 + r;                            // replicate-pad clamp
        grow = grow < 0 ? 0 : (grow > NXc - 1 ? NXc - 1 : grow);
        const float* src = sel ? (perm + base0) : (pressure + base);
        const uint64_t gaddr = (uint64_t)(uintptr_t)(src + grow * NYc + lane * 4);
        float* dstp = (sel ? sK0 : sP) + r * NYc + lane * 4;
        const uint32_t loff = (uint32_t)(uintptr_t)dstp;  // low 32 bits = LDS offset
        asm volatile("global_load_async_to_lds_b128 %0, %1, off"
                     :: "v"(loff), "v"(gaddr) : "memory");
    }
    asm volatile("s_wait_asynccnt 0" ::: "memory");       // this wave's async done
    __syncthreads();                                      // all waves' rows visible

    // ---- Scalar coefficients (t=0 mobilities collapse to constants) ------
    const float sini = Swini[0];                          // siniuse
    const float S0   = (sini - 0.1f) * 1.25f;             // (s-SWI)/(1-SWI-SWR)
    const float Mw0  = S0 * S0;                           // /(UW*BW)=1
    const float Mo0  = (1.f - S0) * (1.f - S0) * (1.f / 2.75f);
    const float c1   = Mw0 + Mo0;                         // dcd* = c1 * d1(a0)
    const float c2   = Mw0;                               // dad* = c2 * d1(a0)

    const float D1U  = 1000.f * 64.f;      // PINI_ALT * 0.5/dxf (dxf = 1/128)
    const float D2U  = 1000.f * 16384.f;   // PINI_ALT / dxf^2
    const float D1A  = 500.f  * 64.f;      // perm->a slope (500) * 0.5/dxf
    const float OUTS = 1e-5f / 128.f;      // dxf * 1e-5
    const float RDTA = 1.f / 6000.f;       // folded into reciprocal of dta

    const int prevbase = ((b * Tc + (t - 1)) * NZc + z) * PLANE;  // valid if t>0

    // ---- Stage 2: fused stencil + pointwise residuals, 4-wide vectorized --
    // lane owns cols [4*lane, 4*lane+3]; wave owns rows wave, wave+8, +16, +24
    const int c0 = lane * 4;
    const int cL = (c0 == 0) ? 0 : c0 - 1;                 // clamped col c0-1
    const int cR = (c0 + 4 > NYc - 1) ? NYc - 1 : c0 + 4;  // clamped col c0+4

    for (int r = 0; r < 4; ++r) {
        const int lx = wave + r * 8;                       // local row 0..31
        const int gx = x0 + lx;
        const int rC = lx + 1;
        const int rM = ((gx == 0)       ? 0       : gx - 1) - x0 + 1;
        const int rP = ((gx == NXc - 1) ? NXc - 1 : gx + 1) - x0 + 1;

        const v4f pC  = *(const v4f*)&sP [rC * NYc + c0];
        const v4f pXm = *(const v4f*)&sP [rM * NYc + c0];
        const v4f pXp = *(const v4f*)&sP [rP * NYc + c0];
        const float pl = sP [rC * NYc + cL];
        const float pr = sP [rC * NYc + cR];
        const v4f kC  = *(const v4f*)&sK0[rC * NYc + c0];
        const v4f kXm = *(const v4f*)&sK0[rM * NYc + c0];
        const v4f kXp = *(const v4f*)&sK0[rP * NYc + c0];
        const float kl = sK0[rC * NYc + cL];
        const float kr = sK0[rC * NYc + cR];

        // y-shifted vectors (edge replication handled by cL/cR clamping)
        const v4f pYm = { pl,    pC[0], pC[1], pC[2] };
        const v4f pYp = { pC[1], pC[2], pC[3], pr    };
        const v4f kYm = { kl,    kC[0], kC[1], kC[2] };
        const v4f kYp = { kC[1], kC[2], kC[3], kr    };

        const int idx = base + gx * NYc + c0;
        const v4f q   = __builtin_nontemporal_load((const v4f*)(Q     + idx));
        const v4f qw  = __builtin_nontemporal_load((const v4f*)(Qw    + idx));
        const v4f tm  = __builtin_nontemporal_load((const v4f*)(Timef + idx));
        const v4f phi = __builtin_nontemporal_load((const v4f*)(Phi   + idx));
        const v4f pk  = __builtin_nontemporal_load((const v4f*)(perm  + idx));
        const v4f st  = *(const v4f*)(sat + idx);              // reused by t+1 slabs
        v4f sp;
        if (t == 0) { sp = (v4f){sini, sini, sini, sini}; }
        else        { sp = *(const v4f*)(sat + prevbase + gx * NYc + c0); }

        v4f P, SL;
        #pragma unroll
        for (int j = 0; j < 4; ++j) {
            const float dudx = (pXp[j] - pXm[j]) * D1U;
            const float dudy = (pYp[j] - pYm[j]) * D1U;
            const float ddu  = (pXp[j] + pXm[j] + pYp[j] + pYm[j] - 4.f * pC[j]) * D2U;

            const float d1ax = (kXp[j] - kXm[j]) * D1A;
            const float d1ay = (kYp[j] - kYm[j]) * D1A;

            float dsw = st[j] - sp[j]; dsw = dsw < 0.001f ? 0.001f : dsw;
            const float S   = (sp[j] - 0.1f) * 1.25f;
            const float Mw  = S * S;
            const float Mo  = (1.f - S) * (1.f - S) * (1.f / 2.75f);
            const float a_t = 500.f * pk[j];
            const float a1  = (Mw + Mo) * a_t;
            const float a1w = Mw * a_t;

            const float fin  = q[j]  * 5000.f;
            const float finw = qw[j] * 5000.f;
            // dsw/dta with dta = tm*6000  ->  dsw * rcp(tm) * (1/6000)  (v_rcp_f32)
            const float rtm  = __builtin_amdgcn_rcpf(tm[j]);

            const float grad = d1ax * dudx + d1ay * dudy;
            P[j]  = OUTS * (fin + c1 * grad + a1 * ddu);
            SL[j] = OUTS * (phi[j] * (dsw * rtm * RDTA) - (c2 * grad + a1w * ddu + finw));
        }
        __builtin_nontemporal_store(P,  (v4f*)(p_loss + idx));
        __builtin_nontemporal_store(SL, (v4f*)(s_loss + idx));
    }
}

extern "C" void kernel_launch(void* const* d_in, const int* in_sizes, int n_in,
                              void* d_out, int out_size, void* d_ws, size_t ws_size,
                              hipStream_t stream) {
    (void)in_sizes; (void)n_in; (void)out_size; (void)d_ws; (void)ws_size;
    const float* pressure = (const float*)d_in[0];
    const float* perm     = (const float*)d_in[1];
    const float* Q        = (const float*)d_in[2];
    const float* Qw       = (const float*)d_in[3];
    const float* Timef    = (const float*)d_in[4];
    // d_in[5] = Pini (unused by the reference)
    const float* Phi      = (const float*)d_in[6];
    const float* Swini    = (const float*)d_in[7];
    const float* sat      = (const float*)d_in[8];

    const int N = Bc * Tc * NZc * PLANE;   // 5,242,880 per output
    float* p_loss = (float*)d_out;
    float* s_loss = p_loss + N;

    dim3 grid(Bc * Tc * NZc * NTILE);      // 1280 blocks
    blackoil_pino_kernel<<<grid, 256, 0, stream>>>(
        pressure, perm, Q, Qw, Timef, Phi, Swini, sat, p_loss, s_loss);
}